// SE3Transformer_5686536700079
// MI455X (gfx1250) — compile-verified
//
#include <hip/hip_runtime.h>
#include <math.h>

// SE(3) TFN for MI455X (gfx1250, wave32). Edge-weight GEMMs (emb[E,10] @ W[10,NW])
// are computed per 16-edge tile with V_WMMA_F32_16X16X4_F32 (K padded 10->12),
// staged through LDS, then fused with the CG message-passing math.
// Branch-free weight phase: padded K rows are masked on the A side (zeros),
// so B row indices are clamped instead of predicated -> no exec-mask dances.

typedef __attribute__((ext_vector_type(2))) float v2f;
typedef __attribute__((ext_vector_type(8))) float v8f;

#define SQRT_HALF 0.70710678118654752f
#define INV_SQRT3 0.57735026918962576f
#define CG_A 0.31622776601683794f  /* sqrt(1/10) */
#define CG_B 0.54772255750516611f  /* sqrt(3/10) */
#define CG_G 0.63245553203367587f  /* sqrt(2/5)  */

__device__ __forceinline__ float gauss10(float d, int j) {
  float mu = 0.7f + (1.0f / 9.0f) * (float)j;     // linspace(0.7, 1.7, 10)
  float t = d - mu;
  float r = expf(-50.0f * t * t);                 // 1/(2*0.1^2) = 50
  return (j < 10) ? r : 0.0f;                     // K padding columns -> select, no branch
}

__device__ __forceinline__ float cutoff(float d) {
  float u = d * (1.0f / 1.5f);
  float u2 = u * u, u3 = u2 * u, u6 = u3 * u3;
  float p = 1.0f - 28.0f * u6 + 48.0f * u6 * u - 21.0f * u6 * u2;
  return (u < 1.0f) ? p : 0.0f;
}

__device__ __forceinline__ float ssp(float x) {   // shifted softplus
  float sp = (x > 20.0f) ? x : log1pf(expf(x));
  return sp - 0.69314718055994531f;
}

// Compute weights for a 16-edge tile: lw[edge][0..NW) = emb(edge) @ W + b.
// A (16x12 f32) built from the RBF directly in WMMA layout; B tiles of W are
// 12xNW with rows >= 10 clamped to row 9 (contribution killed by A zeros).
template <int NW>
__device__ __forceinline__ void edge_weights_wmma(
    const float* __restrict__ elen, const float* __restrict__ W,
    const float* __restrict__ bias, float* lw, long e0, long E, int lane) {
  const int rc = lane & 15;   // A-row (edge in tile) == B/D column index
  const int kh = lane >> 4;   // K half: lanes 0-15 -> K {k0,k0+1}, 16-31 -> {k0+2,k0+3}
  long eA = e0 + rc;
  if (eA >= E) eA = E - 1;    // clamp: tail rows produce unused LDS data
  float d = elen[eA];
  v2f a[3];
#pragma unroll
  for (int s = 0; s < 3; ++s) {
    a[s].x = gauss10(d, 4 * s + 2 * kh + 0);
    a[s].y = gauss10(d, 4 * s + 2 * kh + 1);
  }
#pragma unroll
  for (int ct = 0; ct < NW / 16; ++ct) {
    int col = ct * 16 + rc;
    float bb = bias[col];
    v8f acc = {bb, bb, bb, bb, bb, bb, bb, bb};  // all D rows of a lane share one column
#pragma unroll
    for (int s = 0; s < 3; ++s) {
      int k = 4 * s + 2 * kh;
      int ka = (k < 9) ? k : 9;          // clamp padded rows (A side is zero there)
      int kb = (k + 1 < 9) ? k + 1 : 9;
      v2f b;
      b.x = W[(size_t)ka * NW + col];
      b.y = W[(size_t)kb * NW + col];
      acc = __builtin_amdgcn_wmma_f32_16x16x4_f32(false, a[s], false, b,
                                                  (short)0, acc, false, false);
    }
#pragma unroll
    for (int v = 0; v < 8; ++v) {
      // D layout: VGPR v -> M = v + 8*kh, N = col
      lw[(v + 8 * kh) * NW + col] = acc[v];
    }
  }
}

// ---------------- Layer 1: scalar input (N,1) -> (N,4,32) ----------------
__global__ __launch_bounds__(32) void se3_layer1(
    const float* __restrict__ x, const int* __restrict__ eidx,
    const float* __restrict__ esh, const float* __restrict__ elen,
    const float* __restrict__ W1, const float* __restrict__ b1,
    float* __restrict__ hout, int E) {
  __shared__ float lw[16 * 64];
  const int lane = threadIdx.x;
  const long e0 = (long)blockIdx.x * 16;
  edge_weights_wmma<64>(elen, W1, b1, lw, e0, E, lane);
  __syncthreads();
  const int el = lane >> 1;             // edge in tile
  const int ch0 = (lane & 1) * 16;      // channel half
  long e = e0 + el;
  if (e >= E) return;
  int src = eidx[e];
  int dst = eidx[(size_t)E + e];
  const float* sh = esh + (size_t)e * 9;
  float sh0 = sh[0], sh1 = sh[1], sh2 = sh[2], sh3 = sh[3];
  float ew = cutoff(elen[e]);
  float xj = x[src];
  float f0 = xj * sh0 * ew;   // path (0,0,0), CG=1
  float f1 = xj * ew;         // path (0,1,1), CG=identity
  float* outp = hout + (size_t)dst * 128 + ch0;
  const float* wp = lw + el * 64 + ch0;
#pragma unroll
  for (int c = 0; c < 16; ++c) {
    float w0 = wp[c], w1 = wp[32 + c];
    atomicAdd(outp + c, f0 * w0);
    atomicAdd(outp + 32 + c, f1 * sh1 * w1);
    atomicAdd(outp + 64 + c, f1 * sh2 * w1);
    atomicAdd(outp + 96 + c, f1 * sh3 * w1);
  }
}

// ---------------- Layer 2: (N,4,32) -> (N,4,32), 6 CG paths ----------------
__global__ __launch_bounds__(32) void se3_layer2(
    const float* __restrict__ hin, const int* __restrict__ eidx,
    const float* __restrict__ esh, const float* __restrict__ elen,
    const float* __restrict__ W2, const float* __restrict__ b2,
    float* __restrict__ hout, int E) {
  __shared__ float lw[16 * 192];
  const int lane = threadIdx.x;
  const long e0 = (long)blockIdx.x * 16;
  edge_weights_wmma<192>(elen, W2, b2, lw, e0, E, lane);
  __syncthreads();
  const int el = lane >> 1;
  const int ch0 = (lane & 1) * 16;
  long e = e0 + el;
  if (e >= E) return;
  int src = eidx[e];
  int dst = eidx[(size_t)E + e];
  const float* sh = esh + (size_t)e * 9;
  float sh0 = sh[0], sh1 = sh[1], sh2 = sh[2], sh3 = sh[3];
  float q0 = sh[4], q1 = sh[5], q2 = sh[6], q3 = sh[7], q4 = sh[8];
  float ew = cutoff(elen[e]);
  const float* inp = hin + (size_t)src * 128 + ch0;
  float* outp = hout + (size_t)dst * 128 + ch0;
  const float* wp = lw + el * 192 + ch0;
  // Real CG(1,2,1): symmetric traceless quadrupole action, M[a][k]
  float m00 = CG_A * q2 + CG_B * q4, m10 = -CG_B * q1, m20 = -CG_B * q0;
  float m01 = -CG_B * q1, m11 = -CG_G * q2, m21 = -CG_B * q3;
  float m02 = -CG_B * q0, m12 = -CG_B * q3, m22 = CG_A * q2 - CG_B * q4;
#pragma unroll 4
  for (int c = 0; c < 16; ++c) {
    float h0 = inp[c];
    float v0 = inp[32 + c], v1 = inp[64 + c], v2 = inp[96 + c];
    float w0 = wp[c], w1 = wp[32 + c], w2 = wp[64 + c];
    float w3 = wp[96 + c], w4 = wp[128 + c], w5 = wp[160 + c];
    float dot = v0 * sh1 + v1 * sh2 + v2 * sh3;        // path (1,1,0): -dot/sqrt3
    float cr0 = v1 * sh3 - v2 * sh2;                   // path (1,1,1): -(v x s)/sqrt2
    float cr1 = v2 * sh1 - v0 * sh3;
    float cr2 = v0 * sh2 - v1 * sh1;
    float t50 = v0 * m00 + v1 * m10 + v2 * m20;        // path (1,2,1)
    float t51 = v0 * m01 + v1 * m11 + v2 * m21;
    float t52 = v0 * m02 + v1 * m12 + v2 * m22;
    float o0 = (h0 * sh0 * w0 - INV_SQRT3 * dot * w3) * ew;
    float o1 = (h0 * sh1 * w1 + v0 * sh0 * w2 - SQRT_HALF * cr0 * w4 + t50 * w5) * ew;
    float o2 = (h0 * sh2 * w1 + v1 * sh0 * w2 - SQRT_HALF * cr1 * w4 + t51 * w5) * ew;
    float o3 = (h0 * sh3 * w1 + v2 * sh0 * w2 - SQRT_HALF * cr2 * w4 + t52 * w5) * ew;
    atomicAdd(outp + c, o0);
    atomicAdd(outp + 32 + c, o1);
    atomicAdd(outp + 64 + c, o2);
    atomicAdd(outp + 96 + c, o3);
  }
}

// ---------------- Layer 3: (N,4,32) -> (N,1,32) ----------------
__global__ __launch_bounds__(32) void se3_layer3(
    const float* __restrict__ hin, const int* __restrict__ eidx,
    const float* __restrict__ esh, const float* __restrict__ elen,
    const float* __restrict__ W3, const float* __restrict__ b3,
    float* __restrict__ hout, int E) {
  __shared__ float lw[16 * 64];
  const int lane = threadIdx.x;
  const long e0 = (long)blockIdx.x * 16;
  edge_weights_wmma<64>(elen, W3, b3, lw, e0, E, lane);
  __syncthreads();
  const int el = lane >> 1;
  const int ch0 = (lane & 1) * 16;
  long e = e0 + el;
  if (e >= E) return;
  int src = eidx[e];
  int dst = eidx[(size_t)E + e];
  const float* sh = esh + (size_t)e * 9;
  float sh0 = sh[0], sh1 = sh[1], sh2 = sh[2], sh3 = sh[3];
  float ew = cutoff(elen[e]);
  const float* inp = hin + (size_t)src * 128 + ch0;
  float* outp = hout + (size_t)dst * 32 + ch0;
  const float* wp = lw + el * 64 + ch0;
#pragma unroll
  for (int c = 0; c < 16; ++c) {
    float h0 = inp[c];
    float v0 = inp[32 + c], v1 = inp[64 + c], v2 = inp[96 + c];
    float dot = v0 * sh1 + v1 * sh2 + v2 * sh3;
    float o0 = (h0 * sh0 * wp[c] - INV_SQRT3 * dot * wp[32 + c]) * ew;
    atomicAdd(outp + c, o0);
  }
}

// ---------------- Pointwise nonlinearities ----------------
__global__ void se3_norm_act(float* __restrict__ h, int N) {
  long i = (long)blockIdx.x * blockDim.x + threadIdx.x;
  if (i >= (long)N * 32) return;
  long n = i >> 5;
  int c = (int)(i & 31);
  float* p = h + n * 128 + c;
  float s = p[0], a = p[32], b = p[64], d = p[96];
  float nv = sqrtf(a * a + b * b + d * d + 1e-12f);
  float sc = ssp(nv) / nv;
  p[0] = ssp(s);
  p[32] = a * sc;
  p[64] = b * sc;
  p[96] = d * sc;
}

__global__ void se3_silu(float* __restrict__ h, int N) {
  long i = (long)blockIdx.x * blockDim.x + threadIdx.x;
  if (i >= (long)N * 32) return;
  float v = h[i];
  h[i] = v / (1.0f + expf(-v));
}

// ---------------- Graph pooling (batch is sorted) ----------------
__global__ void se3_reduce(const float* __restrict__ h3,
                           const int* __restrict__ batch,
                           float* __restrict__ g, int N) {
  int tid = blockIdx.x * blockDim.x + threadIdx.x;  // 1024 chunks x 32 channels
  int c = tid & 31;
  int chunk = tid >> 5;
  const int P = 1024;
  if (chunk >= P) return;
  int per = (N + P - 1) / P;
  int nb = chunk * per;
  int ne = (nb + per < N) ? nb + per : N;
  float acc = 0.0f;
  int cur = -1;
  for (int n = nb; n < ne; ++n) {
    int bi = batch[n];
    if (bi != cur) {
      if (cur >= 0) atomicAdd(&g[(size_t)cur * 32 + c], acc);
      cur = bi;
      acc = 0.0f;
    }
    acc += h3[(size_t)n * 32 + c];
  }
  if (cur >= 0) atomicAdd(&g[(size_t)cur * 32 + c], acc);
}

// ---------------- Head: softmax(g @ Wo + bo) ----------------
__global__ void se3_head(const float* __restrict__ g, const float* __restrict__ Wo,
                         const float* __restrict__ bo, float* __restrict__ out) {
  int b = blockIdx.x * blockDim.x + threadIdx.x;
  if (b >= 128) return;
  float lg[8];
#pragma unroll
  for (int o = 0; o < 8; ++o) {
    float s = bo[o];
    for (int c = 0; c < 32; ++c) s += g[(size_t)b * 32 + c] * Wo[(size_t)c * 8 + o];
    lg[o] = s;
  }
  float mx = lg[0];
#pragma unroll
  for (int o = 1; o < 8; ++o) mx = fmaxf(mx, lg[o]);
  float se = 0.0f;
#pragma unroll
  for (int o = 0; o < 8; ++o) {
    lg[o] = expf(lg[o] - mx);
    se += lg[o];
  }
  float inv = 1.0f / se;
#pragma unroll
  for (int o = 0; o < 8; ++o) out[(size_t)b * 8 + o] = lg[o] * inv;
}

extern "C" void kernel_launch(void* const* d_in, const int* in_sizes, int n_in,
                              void* d_out, int out_size, void* d_ws, size_t ws_size,
                              hipStream_t stream) {
  (void)n_in; (void)out_size; (void)ws_size;
  const float* x    = (const float*)d_in[0];
  const int*   eidx = (const int*)d_in[1];
  const float* esh  = (const float*)d_in[2];
  const float* elen = (const float*)d_in[3];
  const int*   batch= (const int*)d_in[4];
  const float* W1 = (const float*)d_in[5];
  const float* b1 = (const float*)d_in[6];
  const float* W2 = (const float*)d_in[7];
  const float* b2 = (const float*)d_in[8];
  const float* W3 = (const float*)d_in[9];
  const float* b3 = (const float*)d_in[10];
  const float* Wo = (const float*)d_in[11];
  const float* bo = (const float*)d_in[12];
  const int N = in_sizes[0];   // x is (N,1)
  const int E = in_sizes[3];   // edge_len is (E,)

  float* hA = (float*)d_ws;                     // N*128
  float* hB = hA + (size_t)N * 128;             // N*128
  float* h3 = hB + (size_t)N * 128;             // N*32
  float* g  = h3 + (size_t)N * 32;              // 128*32

  const int tiles = (E + 15) / 16;
  const int na = (N * 32 + 255) / 256;

  hipMemsetAsync(hA, 0, (size_t)N * 128 * sizeof(float), stream);
  se3_layer1<<<tiles, 32, 0, stream>>>(x, eidx, esh, elen, W1, b1, hA, E);
  se3_norm_act<<<na, 256, 0, stream>>>(hA, N);

  hipMemsetAsync(hB, 0, (size_t)N * 128 * sizeof(float), stream);
  se3_layer2<<<tiles, 32, 0, stream>>>(hA, eidx, esh, elen, W2, b2, hB, E);
  se3_norm_act<<<na, 256, 0, stream>>>(hB, N);

  hipMemsetAsync(h3, 0, (size_t)N * 32 * sizeof(float), stream);
  se3_layer3<<<tiles, 32, 0, stream>>>(hB, eidx, esh, elen, W3, b3, h3, E);
  se3_silu<<<na, 256, 0, stream>>>(h3, N);

  hipMemsetAsync(g, 0, 128 * 32 * sizeof(float), stream);
  se3_reduce<<<128, 256, 0, stream>>>(h3, batch, g, N);
  se3_head<<<1, 128, 0, stream>>>(g, Wo, bo, (float*)d_out);
}